// Orthogonal2_71528385347988
// MI455X (gfx1250) — compile-verified
//
#include <hip/hip_runtime.h>

typedef __attribute__((ext_vector_type(2))) float v2f;
typedef __attribute__((ext_vector_type(8))) float v8f;

#define NDIM 2048
#define BM 64
#define BN 256
#define BK 16
#define SQUARINGS 12
#define TAYLOR_MAX 8

// LDS layout (floats), one combined array:
//   per buffer: A tile 64x16 padded to stride 20  -> 1280 floats
//               B tile 16x256 padded to stride 260 -> 4160 floats
#define A_STRIDE 20
#define B_STRIDE 260
#define ASZ (BM * A_STRIDE)          // 1280
#define BSZ (BK * B_STRIDE)          // 4160
#define BUFSZ (ASZ + BSZ)            // 5440 floats
#define BUFBYTES (BUFSZ * 4)         // 21760 bytes (16B aligned)

// ---------------------------------------------------------------------------
// A = scale * (triu(L,1) - triu(L,1)^T)
// ---------------------------------------------------------------------------
__global__ __launch_bounds__(256) void build_skew_kernel(
    const float* __restrict__ L, float* __restrict__ As, float scale) {
  int idx = blockIdx.x * blockDim.x + threadIdx.x;
  int i = idx >> 11;       // NDIM = 2048
  int j = idx & (NDIM - 1);
  float u = (j > i) ? L[(size_t)i * NDIM + j] : 0.0f;
  float l = (i > j) ? L[(size_t)j * NDIM + i] : 0.0f;
  As[idx] = (u - l) * scale;
}

// ---------------------------------------------------------------------------
// S = I + As   (Taylor terms k=0,1),   T = As  (current term)
// ---------------------------------------------------------------------------
__global__ __launch_bounds__(256) void init_ST_kernel(
    const float* __restrict__ As, float* __restrict__ S, float* __restrict__ T) {
  int idx = blockIdx.x * blockDim.x + threadIdx.x;
  int i = idx >> 11;
  int j = idx & (NDIM - 1);
  float a = As[idx];
  T[idx] = a;
  S[idx] = a + ((i == j) ? 1.0f : 0.0f);
}

// Async global->LDS b128. lds = low 32 bits of the flat address of the LDS
// destination (== wave-relative LDS byte address); ga = 64-bit global address.
// The unreferenced %2 operand escapes `smem` into the asm so the "memory"
// clobber forces the compiler to treat LDS as written (keeps ds_loads alive).
#define ASYNC_B128(lds, ga, base)                                         \
  asm volatile("global_load_async_to_lds_b128 %0, %1, off"                \
               :: "v"(lds), "v"(ga), "s"(base) : "memory")
#define ASYNC_B128_OFF(lds, ga, base, off)                                \
  asm volatile("global_load_async_to_lds_b128 %0, %1, off offset:" #off   \
               :: "v"(lds), "v"(ga), "s"(base) : "memory")

// ---------------------------------------------------------------------------
// C[M x N] = alpha * A[M x K] @ B[K x ldb][:, :N]; optionally S += C.
// 256 threads = 8 wave32; block tile 64x256; wave tile 32x64 (2x4 WMMA frags).
// Global -> LDS staging uses CDNA5 async-to-LDS (ASYNCcnt), double-buffered.
// M % 64 == 0, N % 256 == 0, K % 16 == 0 for all call sites -> EXEC all-ones.
// ---------------------------------------------------------------------------
__global__ __launch_bounds__(256) void wmma_gemm_f32_kernel(
    const float* __restrict__ A, const float* __restrict__ B,
    float* __restrict__ C, float* __restrict__ S,
    int M, int N, int K, int ldb, float alpha) {
  __shared__ float smem[2 * BUFSZ];
  (void)M;

  const int tid  = threadIdx.x;
  const int lane = tid & 31;
  const int wave = tid >> 5;
  const int wr   = wave & 1;      // 2 waves down  -> 64 rows
  const int wc   = wave >> 1;     // 4 waves across-> 4*64 = 256 cols
  const int m0   = blockIdx.y * BM;
  const int n0   = blockIdx.x * BN;

  const int lhalf = lane >> 4;    // 0: lanes 0-15, 1: lanes 16-31
  const int lmod  = lane & 15;

  // Staging slices: A -> 1 x b128 per thread; B -> 4 x b128 (16-float row run)
  const int ar = tid >> 2;            // 0..63
  const int ac = (tid & 3) * 4;       // 0,4,8,12
  const int br = tid >> 4;            // 0..15
  const int bc = (tid & 15) * 16;     // 0..240

  const unsigned long long gA0 =
      (unsigned long long)(const void*)(A + (size_t)(m0 + ar) * K + ac);
  const unsigned long long gB0 =
      (unsigned long long)(const void*)(B + (size_t)br * ldb + n0 + bc);

  // LDS byte addresses from the *actual* element pointers (ptrtoint captures
  // smem; low 32 bits of a flat LDS-aperture pointer = LDS byte address).
  const unsigned ldsA0 = (unsigned)(size_t)&smem[ar * A_STRIDE + ac];
  const unsigned ldsB0 = (unsigned)(size_t)&smem[ASZ + br * B_STRIDE + bc];

  v8f c0[4] = {{}, {}, {}, {}};
  v8f c1[4] = {{}, {}, {}, {}};

  const int T = K / BK;

  // Prologue: async-stage tile 0 into buffer 0 (5 ops -> ASYNCcnt)
  ASYNC_B128(ldsA0, gA0, smem);
  ASYNC_B128(ldsB0, gB0, smem);
  ASYNC_B128_OFF(ldsB0, gB0, smem, 16);
  ASYNC_B128_OFF(ldsB0, gB0, smem, 32);
  ASYNC_B128_OFF(ldsB0, gB0, smem, 48);

  int cur = 0;
  for (int t = 0; t < T; ++t) {
    if (t + 1 < T) {
      // Issue next tile into the other buffer, then wait for the current tile
      // (<=5 outstanding: everything older than the 5 just issued is done).
      const int nxt = cur ^ 1;
      const unsigned long long ga = gA0 + (unsigned long long)(t + 1) * BK * 4ull;
      const unsigned long long gb = gB0 + (unsigned long long)(t + 1) * BK * (size_t)ldb * 4ull;
      const unsigned la = ldsA0 + (unsigned)nxt * BUFBYTES;
      const unsigned lb = ldsB0 + (unsigned)nxt * BUFBYTES;
      ASYNC_B128(la, ga, smem);
      ASYNC_B128(lb, gb, smem);
      ASYNC_B128_OFF(lb, gb, smem, 16);
      ASYNC_B128_OFF(lb, gb, smem, 32);
      ASYNC_B128_OFF(lb, gb, smem, 48);
      asm volatile("s_wait_asynccnt 0x5" ::: "memory");
    } else {
      asm volatile("s_wait_asynccnt 0x0" ::: "memory");
    }
    __syncthreads();   // all waves' slices of tile t resident in buf[cur]

    const float* Ab = smem + cur * BUFSZ;
    const float* Bb = Ab + ASZ;
#pragma unroll
    for (int kk = 0; kk < BK; kk += 4) {
      // A-frag 16x4 f32: lanes 0-15 hold K=kk,kk+1; lanes 16-31 hold K=kk+2,kk+3
      const int kc = kk + 2 * lhalf;
      const v2f a0 = *(const v2f*)&Ab[(wr * 32 + lmod) * A_STRIDE + kc];
      const v2f a1 = *(const v2f*)&Ab[(wr * 32 + 16 + lmod) * A_STRIDE + kc];
#pragma unroll
      for (int j = 0; j < 4; ++j) {
        const int nl = wc * 64 + j * 16 + lmod;
        v2f b;
        b.x = Bb[(kc + 0) * B_STRIDE + nl];
        b.y = Bb[(kc + 1) * B_STRIDE + nl];
        c0[j] = __builtin_amdgcn_wmma_f32_16x16x4_f32(false, a0, false, b, (short)0, c0[j],
                                                      false, false);
        c1[j] = __builtin_amdgcn_wmma_f32_16x16x4_f32(false, a1, false, b, (short)0, c1[j],
                                                      false, false);
      }
    }
    __syncthreads();   // buf[cur^1]'s previous readers done before it is refilled
    cur ^= 1;
  }

  // C/D layout: VGPR r -> lanes 0-15: (M=r, N=lane), lanes 16-31: (M=r+8, N=lane-16)
  const int crow = m0 + wr * 32;
  const int ccol = n0 + wc * 64;
#pragma unroll
  for (int r = 0; r < 8; ++r) {
    const int mr = r + 8 * lhalf;
#pragma unroll
    for (int j = 0; j < 4; ++j) {
      const size_t i0 = (size_t)(crow + mr) * N + ccol + j * 16 + lmod;
      const size_t i1 = (size_t)(crow + 16 + mr) * N + ccol + j * 16 + lmod;
      const float v0 = alpha * c0[j][r];
      const float v1 = alpha * c1[j][r];
      C[i0] = v0;
      C[i1] = v1;
      if (S != nullptr) {   // fused Taylor-series accumulation (disjoint indices)
        S[i0] += v0;
        S[i1] += v1;
      }
    }
  }
}

// ---------------------------------------------------------------------------
// Pipeline: As = skew(L)/2^12 ; E = expm via Taylor(8) + 12 squarings ;
// out = x @ E[:, :1024]
// ---------------------------------------------------------------------------
extern "C" void kernel_launch(void* const* d_in, const int* in_sizes, int n_in,
                              void* d_out, int out_size, void* d_ws, size_t ws_size,
                              hipStream_t stream) {
  (void)in_sizes; (void)n_in; (void)out_size; (void)ws_size;
  const float* x = (const float*)d_in[0];
  const float* L = (const float*)d_in[1];
  float* out = (float*)d_out;

  const size_t NN = (size_t)NDIM * NDIM;
  float* As = (float*)d_ws;
  float* T0 = As + NN;
  float* T1 = T0 + NN;
  float* S  = T1 + NN;

  const float scale = 1.0f / (float)(1 << SQUARINGS);
  const int eth = 256;
  const int ebl = (int)(NN / eth);

  build_skew_kernel<<<ebl, eth, 0, stream>>>(L, As, scale);
  init_ST_kernel<<<ebl, eth, 0, stream>>>(As, S, T0);

  dim3 blk(256);
  dim3 gsq(NDIM / BN, NDIM / BM);   // (8, 32)

  // Taylor terms k = 2..8:  T_k = (1/k) * As @ T_{k-1};  S += T_k
  float* Tp = T0;
  float* Tn = T1;
  for (int k = 2; k <= TAYLOR_MAX; ++k) {
    wmma_gemm_f32_kernel<<<gsq, blk, 0, stream>>>(As, Tp, Tn, S, NDIM, NDIM, NDIM, NDIM,
                                                  1.0f / (float)k);
    float* t = Tp; Tp = Tn; Tn = t;
  }

  // 12 squarings: E <- E @ E (ping-pong; `src` tracks the latest result)
  float* src = S;
  float* dst = T0;
  for (int i = 0; i < SQUARINGS; ++i) {
    wmma_gemm_f32_kernel<<<gsq, blk, 0, stream>>>(src, src, dst, nullptr, NDIM, NDIM, NDIM, NDIM,
                                                  1.0f);
    float* t = src; src = dst; dst = t;
  }

  // out[16384 x 1024] = x[16384 x 2048] @ E[:2048, :1024]  (ldb = 2048)
  dim3 gf(1024 / BN, 16384 / BM);   // (4, 256)
  wmma_gemm_f32_kernel<<<gf, blk, 0, stream>>>(x, src, out, nullptr, 16384, 1024, NDIM, NDIM,
                                               1.0f);
}